// Denoising_Model_24764781429260
// MI455X (gfx1250) — compile-verified
//
#include <hip/hip_runtime.h>
#include <math.h>

typedef __attribute__((ext_vector_type(2))) float v2f;
typedef __attribute__((ext_vector_type(8))) float v8f;

#define HDIM 128

// ---------------------------------------------------------------- utilities
__global__ void zero_f32(float* __restrict__ p, int n) {
  int i = blockIdx.x * blockDim.x + threadIdx.x;
  if (i < n) p[i] = 0.f;
}

// interleave 128x128 weight: Wp[(k>>1)*256 + j*2 + (k&1)] = W[k*128+j]
__global__ void interleave_wn(const float* __restrict__ W, float* __restrict__ Wp) {
  int idx = blockIdx.x * blockDim.x + threadIdx.x;
  if (idx >= 128 * 128) return;
  int k = idx >> 7, j = idx & 127;
  Wp[(k >> 1) * 256 + j * 2 + (k & 1)] = W[idx];
}

// zero-pad + interleave Wd1 [257,257] -> [130 pair-rows][272 cols][2]
__global__ void pad_wd1(const float* __restrict__ Wd1, float* __restrict__ Wp) {
  int idx = blockIdx.x * blockDim.x + threadIdx.x;
  if (idx >= 130 * 544) return;
  int p = idx / 544, rem = idx - p * 544;
  int j = rem >> 1, b = rem & 1;
  int k = p * 2 + b;
  Wp[idx] = (k < 257 && j < 257) ? Wd1[k * 257 + j] : 0.f;
}

__global__ void pad_small(const float* __restrict__ bd1, const float* __restrict__ Wd2,
                          float* __restrict__ bd1p, float* __restrict__ Wd2p) {
  int i = threadIdx.x;
  if (i < 272) {
    bd1p[i] = (i < 257) ? bd1[i] : 0.f;
    Wd2p[i] = (i < 257) ? Wd2[i] : 0.f;
  }
}

// ------------------------------------------------------- tiny setup (1 block)
__global__ void setup_small(const float* __restrict__ t,
                            const float* __restrict__ Wt1, const float* __restrict__ bt1,
                            const float* __restrict__ Wt2, const float* __restrict__ bt2,
                            const float* __restrict__ We,  const float* __restrict__ be,
                            const float* __restrict__ Wo,  const float* __restrict__ bo,
                            const float* __restrict__ Wl1, const float* __restrict__ bl1,
                            const float* __restrict__ Wl2, const float* __restrict__ bl2,
                            const float* __restrict__ Wl3, const float* __restrict__ bl3,
                            float* __restrict__ temb, float* __restrict__ uvw1,
                            float* __restrict__ uvw2, float* __restrict__ uvw3) {
  __shared__ float temb0[128];
  __shared__ float amid[256];
  const int tid = threadIdx.x;  // 256 threads

  if (tid < 64) {
    float f  = expf((float)tid * (-logf(10000.f) / 63.f));
    float tv = t[0] * 4.f * f;
    temb0[tid]      = sinf(tv);
    temb0[64 + tid] = cosf(tv);
  }
  __syncthreads();
  {
    float s = bt1[tid];
    for (int k = 0; k < 128; ++k) s += temb0[k] * Wt1[k * 256 + tid];
    amid[tid] = fmaxf(s, 0.f);
  }
  __syncthreads();
  if (tid < 128) {
    float s = bt2[tid];
    for (int k = 0; k < 256; ++k) s += amid[k] * Wt2[k * 128 + tid];
    temb[tid] = s;
  }
  if (tid == 0) {  // layer 1: Wl1 is [128,1]
    float u = 0.f, v = 0.f, w = bl1[0];
    for (int k = 0; k < 64; ++k) {
      u += We[k] * Wl1[k];
      v += Wo[k] * Wl1[64 + k];
      w += be[k] * Wl1[k] + bo[k] * Wl1[64 + k];
    }
    uvw1[0] = u; uvw1[1] = v; uvw1[2] = w;
  }
  {  // layers 2 and 3
    const float* Wl = (tid < 128) ? Wl2 : Wl3;
    const float* bl = (tid < 128) ? bl2 : bl3;
    float* dstv     = (tid < 128) ? uvw2 : uvw3;
    int j = tid & 127;
    float u = 0.f, v = 0.f, w = bl[j];
    for (int k = 0; k < 64; ++k) {
      float wlo = Wl[k * HDIM + j];
      float whi = Wl[(64 + k) * HDIM + j];
      u += We[k] * wlo;
      v += Wo[k] * whi;
      w += be[k] * wlo + bo[k] * whi;
    }
    dstv[j] = u; dstv[128 + j] = v; dstv[256 + j] = w;
  }
}

// ----------------------------------------------------------- layer 1 (d_in=1)
__global__ void l1_edge(const float* __restrict__ x, const float* __restrict__ ea,
                        const float* __restrict__ q, const int* __restrict__ src,
                        const int* __restrict__ dst, const float* __restrict__ uvw1,
                        float* agg1, int E) {
  int e = blockIdx.x * blockDim.x + threadIdx.x;
  if (e >= E) return;
  float m = x[src[e]] + ea[e] * uvw1[0] + q[e] * uvw1[1] + uvw1[2];
  atomicAdd(&agg1[dst[e]], fmaxf(m, 0.f));
}

__global__ void l1_node(const float* __restrict__ x, const float* __restrict__ agg1,
                        const float* __restrict__ Wn1, const float* __restrict__ bn1,
                        const float* __restrict__ temb, float* __restrict__ h, int N) {
  int gid = blockIdx.x * blockDim.x + threadIdx.x;
  int n = gid >> 7, j = gid & 127;
  if (n >= N) return;
  h[gid] = (x[n] + agg1[n]) * Wn1[j] + bn1[j] + temb[j];
}

// --------------------------------------- fused edge message + relu + scatter
// one wave per edge, float4 per lane
__global__ void edge_scatter(const float* __restrict__ h, const float* __restrict__ ea,
                             const float* __restrict__ q, const int* __restrict__ src,
                             const int* __restrict__ dst, const float* __restrict__ uvw,
                             float* agg, int E) {
  int t = blockIdx.x * blockDim.x + threadIdx.x;
  int e = t >> 5;
  if (e >= E) return;
  int c4 = (t & 31) * 4;
  int sb = src[e] * HDIM, db = dst[e] * HDIM;
  float eav = ea[e], qv = q[e];
  const float4 hv = *(const float4*)&h[sb + c4];
  const float4 uv = *(const float4*)&uvw[c4];
  const float4 vv = *(const float4*)&uvw[128 + c4];
  const float4 wv = *(const float4*)&uvw[256 + c4];
  atomicAdd(&agg[db + c4 + 0], fmaxf(hv.x + eav * uv.x + qv * vv.x + wv.x, 0.f));
  atomicAdd(&agg[db + c4 + 1], fmaxf(hv.y + eav * uv.y + qv * vv.y + wv.y, 0.f));
  atomicAdd(&agg[db + c4 + 2], fmaxf(hv.z + eav * uv.z + qv * vv.z + wv.z, 0.f));
  atomicAdd(&agg[db + c4 + 3], fmaxf(hv.w + eav * uv.w + qv * vv.w + wv.w, 0.f));
}

// ----------------------------- node update GEMM: out = (h+agg)@Wn + bn + temb
// 256 threads (8 waves), 32-row tile; wave = col-tile, 2 row-tile accumulators.
// Wnp is the k-pair interleaved weight. `out` may alias `agg` (staged via LDS).
#define ASTRIDE 132
__global__ void node_gemm(const float* __restrict__ h, const float* agg,
                          const float* __restrict__ Wnp, const float* __restrict__ bn,
                          const float* __restrict__ temb, float* out, int N) {
  __shared__ float A[32 * ASTRIDE];
  const int tid  = threadIdx.x;
  const int lane = tid & 31;
  const int wave = tid >> 5;
  const int m0   = blockIdx.x * 32;

  for (int idx = tid; idx < 32 * 128; idx += 256) {
    int r = idx >> 7, c = idx & 127;
    int row = m0 + r; if (row >= N) row = N - 1;
    A[r * ASTRIDE + c] = h[row * HDIM + c] + agg[row * HDIM + c];
  }
  __syncthreads();

  const int am   = lane & 15;
  const int koff = (lane >> 4) << 1;                 // 0 or 2
  const int col  = wave * 16 + (lane & 15);
  const float* bp = Wnp + (lane >> 4) * 256 + col * 2;  // k-pair row stride = 256
  v8f acc0 = {}, acc1 = {};
#pragma unroll 8
  for (int kk = 0; kk < 128; kk += 4) {
    const int ka = kk + koff;
    const v2f b  = *(const v2f*)bp;
    const v2f a0 = *(const v2f*)&A[am * ASTRIDE + ka];
    const v2f a1 = *(const v2f*)&A[(16 + am) * ASTRIDE + ka];
    acc0 = __builtin_amdgcn_wmma_f32_16x16x4_f32(false, a0, false, b, (short)0, acc0,
                                                 false, false);
    acc1 = __builtin_amdgcn_wmma_f32_16x16x4_f32(false, a1, false, b, (short)0, acc1,
                                                 false, false);
    bp += 512;                                        // 2 pair-rows ahead
  }
  const float add = bn[col] + temb[col];
  const int mb = (lane >> 4) << 3;
#pragma unroll
  for (int r = 0; r < 8; ++r) {
    int row0 = m0 + mb + r;
    int row1 = m0 + 16 + mb + r;
    if (row0 < N) out[row0 * HDIM + col] = acc0[r] + add;
    if (row1 < N) out[row1 * HDIM + col] = acc1[r] + add;
  }
}

// ---------------- decode: out = relu([h_src|h_dst|q]@Wd1 + bd1)@Wd2 + bd2
// 32-edge tile, 256 threads (8 waves). Wd1p padded+interleaved [130][272*2].
// Second matmul fused via shfl reduction + LDS atomics (no Y buffer).
#define ZSTRIDE 276
__global__ void decode_kernel(const float* __restrict__ h, const float* __restrict__ q,
                              const int* __restrict__ src, const int* __restrict__ dst,
                              const float* __restrict__ Wd1p, const float* __restrict__ bd1p,
                              const float* __restrict__ Wd2p, const float* __restrict__ bd2,
                              float* __restrict__ out, int E) {
  __shared__ float Z[32 * ZSTRIDE];
  __shared__ float outAcc[32];
  const int tid  = threadIdx.x;  // 256
  const int lane = tid & 31;
  const int wave = tid >> 5;
  const int e0   = blockIdx.x * 32;

  if (tid < 32) outAcc[tid] = 0.f;
  for (int idx = tid; idx < 32 * 272; idx += 256) {
    int r = idx / 272, k = idx - r * 272;
    int e = e0 + r; if (e >= E) e = E - 1;
    float val;
    if (k < 128)       val = h[src[e] * HDIM + k];
    else if (k < 256)  val = h[dst[e] * HDIM + (k - 128)];
    else if (k == 256) val = q[e];
    else               val = 0.f;
    Z[r * ZSTRIDE + k] = val;
  }
  __syncthreads();

  const int am   = lane & 15;
  const int koff = (lane >> 4) << 1;
  const int mb   = (lane >> 4) << 3;

  for (int tile = wave; tile < 17; tile += 8) {
    const int col = tile * 16 + (lane & 15);
    const float* bp = Wd1p + (lane >> 4) * 544 + col * 2;  // k-pair row stride = 544
    v8f acc0 = {}, acc1 = {};
#pragma unroll 5
    for (int kk = 0; kk < 260; kk += 4) {
      const int ka = kk + koff;
      const v2f b  = *(const v2f*)bp;
      const v2f a0 = *(const v2f*)&Z[am * ZSTRIDE + ka];
      const v2f a1 = *(const v2f*)&Z[(16 + am) * ZSTRIDE + ka];
      acc0 = __builtin_amdgcn_wmma_f32_16x16x4_f32(false, a0, false, b, (short)0, acc0,
                                                   false, false);
      acc1 = __builtin_amdgcn_wmma_f32_16x16x4_f32(false, a1, false, b, (short)0, acc1,
                                                   false, false);
      bp += 1088;
    }
    // fused second matmul: y = relu(acc + bd1) * Wd2 ; reduce over 16 col-lanes
    const float bb   = bd1p[col];
    const float wd2c = Wd2p[col];   // zero for padded cols -> they contribute nothing
    float red0[8], red1[8];
#pragma unroll
    for (int r = 0; r < 8; ++r) {
      float y0 = fmaxf(acc0[r] + bb, 0.f) * wd2c;
      float y1 = fmaxf(acc1[r] + bb, 0.f) * wd2c;
      y0 += __shfl_xor(y0, 1); y0 += __shfl_xor(y0, 2);
      y0 += __shfl_xor(y0, 4); y0 += __shfl_xor(y0, 8);
      y1 += __shfl_xor(y1, 1); y1 += __shfl_xor(y1, 2);
      y1 += __shfl_xor(y1, 4); y1 += __shfl_xor(y1, 8);
      red0[r] = y0; red1[r] = y1;
    }
    if ((lane & 15) == 0) {
#pragma unroll
      for (int r = 0; r < 8; ++r) {
        atomicAdd(&outAcc[mb + r], red0[r]);
        atomicAdd(&outAcc[16 + mb + r], red1[r]);
      }
    }
  }
  __syncthreads();
  if (tid < 32) {
    int e = e0 + tid;
    if (e < E) out[e] = outAcc[tid] + bd2[0];
  }
}

// ------------------------------------------------------------------- launch
extern "C" void kernel_launch(void* const* d_in, const int* in_sizes, int n_in,
                              void* d_out, int out_size, void* d_ws, size_t ws_size,
                              hipStream_t stream) {
  const float* x   = (const float*)d_in[0];
  const float* ea  = (const float*)d_in[1];
  const float* q   = (const float*)d_in[2];
  const int*   adj = (const int*)d_in[3];
  const float* t   = (const float*)d_in[4];
  const float* We  = (const float*)d_in[7];
  const float* be  = (const float*)d_in[8];
  const float* Wo  = (const float*)d_in[9];
  const float* bo  = (const float*)d_in[10];
  const float* Wl1 = (const float*)d_in[11];
  const float* bl1 = (const float*)d_in[12];
  const float* Wn1 = (const float*)d_in[13];
  const float* bn1 = (const float*)d_in[14];
  const float* Wl2 = (const float*)d_in[15];
  const float* bl2 = (const float*)d_in[16];
  const float* Wn2 = (const float*)d_in[17];
  const float* bn2 = (const float*)d_in[18];
  const float* Wl3 = (const float*)d_in[19];
  const float* bl3 = (const float*)d_in[20];
  const float* Wn3 = (const float*)d_in[21];
  const float* bn3 = (const float*)d_in[22];
  const float* Wt1 = (const float*)d_in[23];
  const float* bt1 = (const float*)d_in[24];
  const float* Wt2 = (const float*)d_in[25];
  const float* bt2 = (const float*)d_in[26];
  const float* Wd1 = (const float*)d_in[27];
  const float* bd1 = (const float*)d_in[28];
  const float* Wd2 = (const float*)d_in[29];
  const float* bd2 = (const float*)d_in[30];

  const int N = in_sizes[0];
  const int E = in_sizes[1];
  const int* srcp = adj;
  const int* dstp = adj + E;

  // workspace layout (float offsets, all even -> 8B aligned)
  float* ws    = (float*)d_ws;
  float* temb  = ws;                 // 128
  float* uvw1  = ws + 128;           // 16
  float* uvw2  = ws + 144;           // 384
  float* uvw3  = ws + 528;           // 384
  float* bd1p  = ws + 912;           // 272
  float* Wd2p  = ws + 1184;          // 272
  float* Wn2p  = ws + 1456;          // 16384
  float* Wn3p  = ws + 17840;         // 16384
  float* Wd1p  = ws + 34224;         // 130*544 = 70720
  float* agg1  = ws + 104944;        // N
  size_t off   = (size_t)((104944 + N + 15) & ~15);
  float* buf0  = ws + off;                    // N*128
  float* buf1  = buf0 + (size_t)N * HDIM;     // N*128

  const int NH = N * HDIM;

  // parameter preprocessing
  setup_small<<<1, 256, 0, stream>>>(t, Wt1, bt1, Wt2, bt2, We, be, Wo, bo,
                                     Wl1, bl1, Wl2, bl2, Wl3, bl3,
                                     temb, uvw1, uvw2, uvw3);
  interleave_wn<<<(128 * 128 + 255) / 256, 256, 0, stream>>>(Wn2, Wn2p);
  interleave_wn<<<(128 * 128 + 255) / 256, 256, 0, stream>>>(Wn3, Wn3p);
  pad_wd1<<<(130 * 544 + 255) / 256, 256, 0, stream>>>(Wd1, Wd1p);
  pad_small<<<1, 288, 0, stream>>>(bd1, Wd2, bd1p, Wd2p);

  // layer 1 (scalar features)
  zero_f32<<<(N + 255) / 256, 256, 0, stream>>>(agg1, N);
  l1_edge<<<(E + 255) / 256, 256, 0, stream>>>(x, ea, q, srcp, dstp, uvw1, agg1, E);
  l1_node<<<(NH + 255) / 256, 256, 0, stream>>>(x, agg1, Wn1, bn1, temb, buf0, N);

  // layer 2: h = buf0, agg = buf1, result -> buf1
  zero_f32<<<(NH + 255) / 256, 256, 0, stream>>>(buf1, NH);
  edge_scatter<<<(E * 32 + 255) / 256, 256, 0, stream>>>(buf0, ea, q, srcp, dstp,
                                                         uvw2, buf1, E);
  node_gemm<<<(N + 31) / 32, 256, 0, stream>>>(buf0, buf1, Wn2p, bn2, temb, buf1, N);

  // layer 3: h = buf1, agg = buf0, result -> buf0
  zero_f32<<<(NH + 255) / 256, 256, 0, stream>>>(buf0, NH);
  edge_scatter<<<(E * 32 + 255) / 256, 256, 0, stream>>>(buf1, ea, q, srcp, dstp,
                                                         uvw3, buf0, E);
  node_gemm<<<(N + 31) / 32, 256, 0, stream>>>(buf1, buf0, Wn3p, bn3, temb, buf0, N);

  // decode on edges
  decode_kernel<<<(E + 31) / 32, 256, 0, stream>>>(buf0, q, srcp, dstp, Wd1p, bd1p,
                                                   Wd2p, bd2, (float*)d_out, E);
}